// EMDLosspy_1073741824213
// MI455X (gfx1250) — compile-verified
//
#include <hip/hip_runtime.h>
#include <math.h>

// Problem constants (match reference)
#define BB      8
#define NN      2048
#define EPS_F   0.05f
#define ITERS_F 30
#define LOG_M   (-7.6246189861593985f)     // -log(2048)
#define LN2_F   0.6931471805599453f
// (1/EPS) * log2(e): moves the LSE into base-2 so v_exp_f32/v_log_f32 are used raw
#define SCALE_F 28.853900817779268f

#define ROWS_PER_BLOCK 128
#define THREADS        256                 // 8 waves (wave32), 16 rows per wave

typedef __attribute__((ext_vector_type(2))) float v2f;
typedef __attribute__((ext_vector_type(8))) float v8f;

__device__ __forceinline__ float fast_exp2(float x) { return __builtin_amdgcn_exp2f(x); }
__device__ __forceinline__ float fast_log2(float x) { return __builtin_amdgcn_logf(x); }

// ---------------------------------------------------------------------------
// Sinkhorn half-iteration (log2 domain):
//   dout[m] = EPS * (log_m - ln2 * LSE2_n( (din[n] - C(A_m,B_n)) * SCALE ))
// C recomputed on the fly per 16x16 tile via V_WMMA_F32_16X16X4_F32 (K: 3->4).
// Call (A=template,B=source,din=g,dout=f) for f-update; swapped for g-update.
// ---------------------------------------------------------------------------
__global__ __launch_bounds__(THREADS)
void emd_sinkhorn_update(const float* __restrict__ Apts,
                         const float* __restrict__ Bpts,
                         const float* __restrict__ din,
                         float* __restrict__ dout)
{
    __shared__ float4 sB[NN];              // (x, y, z, |p|^2)  32 KB
    __shared__ float  sg[NN];              // dual-in            8 KB
    __shared__ float4 sA[ROWS_PER_BLOCK];  // this block's rows  2 KB

    const int b        = blockIdx.y;
    const int row_base = blockIdx.x * ROWS_PER_BLOCK;
    const int tid      = threadIdx.x;

    for (int j = tid; j < NN; j += THREADS) {
        const float* p = Bpts + ((size_t)b * NN + j) * 3;
        float x = p[0], y = p[1], z = p[2];
        sB[j] = make_float4(x, y, z, x * x + y * y + z * z);
        sg[j] = din[(size_t)b * NN + j];
    }
    if (tid < ROWS_PER_BLOCK) {
        const int row  = row_base + tid;
        const float* p = Apts + ((size_t)b * NN + row) * 3;
        float x = p[0], y = p[1], z = p[2];
        sA[tid] = make_float4(x, y, z, x * x + y * y + z * z);
    }
    __syncthreads();

    const int lane  = tid & 31;
    const int wave  = tid >> 5;
    const int half  = lane >> 4;   // 0: K=0..1 / M=r   1: K=2..3 / M=8+r
    const int ln    = lane & 15;
    const int wrow0 = wave * 16;

    // A fragment (16x4 f32): lanes 0-15 carry K={0,1}, lanes 16-31 K={2,3}.
    float4 pa = sA[wrow0 + ln];
    v2f afrag;
    afrag.x = half ? pa.z : pa.x;
    afrag.y = half ? 0.0f : pa.y;  // K=3 pad

    float xn[8];
#pragma unroll
    for (int r = 0; r < 8; ++r) xn[r] = sA[wrow0 + 8 * half + r].w;

    float mM[8], sS[8];
#pragma unroll
    for (int r = 0; r < 8; ++r) { mM[r] = -1e30f; sS[r] = 0.0f; }

    for (int ct = 0; ct < NN / 16; ++ct) {
        float4 q = sB[ct * 16 + ln];
        v2f bfrag;                       // B (4x16): same half/K split as A
        bfrag.x = half ? q.z : q.x;
        bfrag.y = half ? 0.0f : q.y;
        const float yn  = q.w;
        const float glS = sg[ct * 16 + ln] * SCALE_F;

        v8f dotc = {};
        dotc = __builtin_amdgcn_wmma_f32_16x16x4_f32(
            false, afrag, false, bfrag, (short)0, dotc, false, false);

#pragma unroll
        for (int r = 0; r < 8; ++r) {
            float C  = fmaxf(fmaf(-2.0f, dotc[r], xn[r] + yn), 0.0f);
            float t  = fmaf(C, -SCALE_F, glS);      // (g - C)/EPS in log2 units
            float mn = fmaxf(mM[r], t);
            sS[r] = fmaf(sS[r], fast_exp2(mM[r] - mn), fast_exp2(t - mn));
            mM[r] = mn;
        }
    }

    // Combine the 16 column-slot partials per row (rows live per-half).
#pragma unroll
    for (int off = 1; off < 16; off <<= 1) {
#pragma unroll
        for (int r = 0; r < 8; ++r) {
            float m2 = __shfl_xor(mM[r], off, 32);
            float s2 = __shfl_xor(sS[r], off, 32);
            float mn = fmaxf(mM[r], m2);
            sS[r] = fmaf(sS[r], fast_exp2(mM[r] - mn),
                         s2 * fast_exp2(m2 - mn));
            mM[r] = mn;
        }
    }

    if (ln == 0) {
#pragma unroll
        for (int r = 0; r < 8; ++r) {
            const int row = row_base + wrow0 + 8 * half + r;
            dout[(size_t)b * NN + row] =
                EPS_F * (LOG_M - LN2_F * (mM[r] + fast_log2(sS[r])));
        }
    }
}

// ---------------------------------------------------------------------------
// Final transport cost: acc[b] += sum_ij exp2((f_i + g_j - C_ij)*SCALE) * C_ij
// ---------------------------------------------------------------------------
__global__ __launch_bounds__(THREADS)
void emd_cost(const float* __restrict__ Apts,
              const float* __restrict__ Bpts,
              const float* __restrict__ f,
              const float* __restrict__ g,
              float* __restrict__ acc)
{
    __shared__ float4 sB[NN];
    __shared__ float  sg[NN];
    __shared__ float4 sA[ROWS_PER_BLOCK];
    __shared__ float  sf[ROWS_PER_BLOCK];

    const int b        = blockIdx.y;
    const int row_base = blockIdx.x * ROWS_PER_BLOCK;
    const int tid      = threadIdx.x;

    for (int j = tid; j < NN; j += THREADS) {
        const float* p = Bpts + ((size_t)b * NN + j) * 3;
        float x = p[0], y = p[1], z = p[2];
        sB[j] = make_float4(x, y, z, x * x + y * y + z * z);
        sg[j] = g[(size_t)b * NN + j];
    }
    if (tid < ROWS_PER_BLOCK) {
        const int row  = row_base + tid;
        const float* p = Apts + ((size_t)b * NN + row) * 3;
        float x = p[0], y = p[1], z = p[2];
        sA[tid] = make_float4(x, y, z, x * x + y * y + z * z);
        sf[tid] = f[(size_t)b * NN + row];
    }
    __syncthreads();

    const int lane  = tid & 31;
    const int wave  = tid >> 5;
    const int half  = lane >> 4;
    const int ln    = lane & 15;
    const int wrow0 = wave * 16;

    float4 pa = sA[wrow0 + ln];
    v2f afrag;
    afrag.x = half ? pa.z : pa.x;
    afrag.y = half ? 0.0f : pa.y;

    float xn[8], ffS[8];
#pragma unroll
    for (int r = 0; r < 8; ++r) {
        xn[r]  = sA[wrow0 + 8 * half + r].w;
        ffS[r] = sf[wrow0 + 8 * half + r] * SCALE_F;
    }

    float partial = 0.0f;
    for (int ct = 0; ct < NN / 16; ++ct) {
        float4 q = sB[ct * 16 + ln];
        v2f bfrag;
        bfrag.x = half ? q.z : q.x;
        bfrag.y = half ? 0.0f : q.y;
        const float yn  = q.w;
        const float glS = sg[ct * 16 + ln] * SCALE_F;

        v8f dotc = {};
        dotc = __builtin_amdgcn_wmma_f32_16x16x4_f32(
            false, afrag, false, bfrag, (short)0, dotc, false, false);

#pragma unroll
        for (int r = 0; r < 8; ++r) {
            float C = fmaxf(fmaf(-2.0f, dotc[r], xn[r] + yn), 0.0f);
            float t = fmaf(C, -SCALE_F, ffS[r] + glS);
            partial = fmaf(fast_exp2(t), C, partial);
        }
    }

#pragma unroll
    for (int off = 1; off < 32; off <<= 1)
        partial += __shfl_xor(partial, off, 32);

    if (lane == 0) atomicAdd(&acc[b], partial);
}

__global__ void emd_zero(float* __restrict__ p, int n)
{
    int i = blockIdx.x * blockDim.x + threadIdx.x;
    if (i < n) p[i] = 0.0f;
}

__global__ void emd_finalize(const float* __restrict__ acc,
                             float* __restrict__ out)
{
    float s = 0.0f;
#pragma unroll
    for (int b = 0; b < BB; ++b) s += acc[b];
    out[0] = s / ((float)BB * (float)NN);   // mean over batch, / n
}

// ---------------------------------------------------------------------------
extern "C" void kernel_launch(void* const* d_in, const int* in_sizes, int n_in,
                              void* d_out, int out_size, void* d_ws, size_t ws_size,
                              hipStream_t stream)
{
    const float* tpl = (const float*)d_in[0];   // template [B,N,3] f32
    const float* src = (const float*)d_in[1];   // source   [B,N,3] f32
    float* out = (float*)d_out;

    float* ws  = (float*)d_ws;
    float* f   = ws;                         // [B*N]
    float* g   = ws + (size_t)BB * NN;       // [B*N]
    float* acc = ws + (size_t)2 * BB * NN;   // [B]

    const int ws_elems = 2 * BB * NN + BB;
    emd_zero<<<(ws_elems + THREADS - 1) / THREADS, THREADS, 0, stream>>>(ws, ws_elems);

    dim3 grid(NN / ROWS_PER_BLOCK, BB);
    for (int it = 0; it < ITERS_F; ++it) {
        emd_sinkhorn_update<<<grid, THREADS, 0, stream>>>(tpl, src, g, f);
        emd_sinkhorn_update<<<grid, THREADS, 0, stream>>>(src, tpl, f, g);
    }

    emd_cost<<<grid, THREADS, 0, stream>>>(tpl, src, f, g, acc);
    emd_finalize<<<1, 1, 0, stream>>>(acc, out);
    (void)in_sizes; (void)n_in; (void)out_size; (void)ws_size;
}